// InteractionMapInit_15942918603418
// MI455X (gfx1250) — compile-verified
//
#include <hip/hip_runtime.h>
#include <hip/hip_bf16.h>
#include <math.h>

// Problem constants (match reference)
#define NR 3200
#define NA 320
#define TD 512
#define DD 128
#define H  128
#define NB 8

typedef float v2f __attribute__((ext_vector_type(2)));
typedef float v4f __attribute__((ext_vector_type(4)));
typedef float v8f __attribute__((ext_vector_type(8)));

// ---------------------------------------------------------------------------
// Kernel 1: init per-batch min/max sentinels
// ---------------------------------------------------------------------------
__global__ void init_minmax_kernel(int* __restrict__ dmin_i, int* __restrict__ dmax_i) {
    int b = threadIdx.x;
    if (b < NB) {
        dmin_i[b] = 0x7F800000;            // +inf bits (positive int)
        dmax_i[b] = (int)0xFF800000u;      // -inf bits (very negative int)
    }
}

// ---------------------------------------------------------------------------
// Kernel 2: C[M,N] = A[M,K] @ W[K,N] + bias[N] using v_wmma_f32_16x16x4_f32.
// One wave computes one 16x16 tile of C, stepping K by 4.
// Fragment layouts per CDNA5 ISA 7.12.2 (32-bit A 16x4, B/C/D row-striped):
//   A: a[v] = A[row0 + (lane&15)][k0 + 2*(lane>>4) + v]
//   B: b[v] = W[k0 + 2*(lane>>4) + v][col0 + (lane&15)]
//   D: d[v] -> C[row0 + v + 8*(lane>>4)][col0 + (lane&15)]
// M,N multiples of 16; K multiple of 4; tiles divisible by 8 waves/block.
// ---------------------------------------------------------------------------
__global__ void gemm_wmma_f32_kernel(const float* __restrict__ A,
                                     const float* __restrict__ W,
                                     const float* __restrict__ bias,
                                     float* __restrict__ C,
                                     int M, int N, int K) {
    const int lane = threadIdx.x & 31;
    const int wave = threadIdx.x >> 5;
    const int tile = blockIdx.x * 8 + wave;
    const int tilesN = N >> 4;
    const int tm = tile / tilesN;
    const int tn = tile - tm * tilesN;
    if (tm >= (M >> 4)) return;

    const int row0 = tm << 4;
    const int col0 = tn << 4;
    const int hl = lane & 15;   // half-lane index
    const int hi = lane >> 4;   // 0 or 1

    v8f acc = {};
    const float* arow = A + (size_t)(row0 + hl) * K;
    for (int k0 = 0; k0 < K; k0 += 4) {
        const int ka = k0 + 2 * hi;
        v2f a, b;
        a.x = arow[ka + 0];
        a.y = arow[ka + 1];
        b.x = W[(size_t)(ka + 0) * N + col0 + hl];
        b.y = W[(size_t)(ka + 1) * N + col0 + hl];
        acc = __builtin_amdgcn_wmma_f32_16x16x4_f32(
            /*neg_a=*/false, a, /*neg_b=*/false, b,
            /*c_mod=*/(short)0, acc, /*reuse_a=*/false, /*reuse_b=*/false);
    }

    const float bv = bias[col0 + hl];
#pragma unroll
    for (int v = 0; v < 8; ++v) {
        C[(size_t)(row0 + v + 8 * hi) * N + col0 + hl] = acc[v] + bv;
    }
}

// ---------------------------------------------------------------------------
// Kernel 3: per-batch masked distance min/max. One thread per residue;
// distances >= 0 so float bit-pattern ordering == int ordering -> int atomics.
// ---------------------------------------------------------------------------
__global__ void minmax_kernel(const float* __restrict__ tpos,
                              const float* __restrict__ dpos,
                              const int* __restrict__ seg_res,
                              const int* __restrict__ seg_atom,
                              int* __restrict__ dmin_i,
                              int* __restrict__ dmax_i) {
    int i = blockIdx.x * blockDim.x + threadIdx.x;
    if (i >= NR) return;
    const int b = seg_res[i];
    const float tx = tpos[3 * i + 0];
    const float ty = tpos[3 * i + 1];
    const float tz = tpos[3 * i + 2];
    float lmin = INFINITY, lmax = -INFINITY;
    bool any = false;
    for (int j = 0; j < NA; ++j) {
        if (seg_atom[j] == b) {
            const float dx = tx - dpos[3 * j + 0];
            const float dy = ty - dpos[3 * j + 1];
            const float dz = tz - dpos[3 * j + 2];
            const float d = sqrtf(dx * dx + dy * dy + dz * dz);
            lmin = fminf(lmin, d);
            lmax = fmaxf(lmax, d);
            any = true;
        }
    }
    if (any) {
        atomicMin(&dmin_i[b], __float_as_int(lmin));
        atomicMax(&dmax_i[b], __float_as_int(lmax));
    }
}

// ---------------------------------------------------------------------------
// Kernel 4: finalize dmin / denom with the reference's fixups.
// ---------------------------------------------------------------------------
__global__ void finalize_kernel(const int* __restrict__ dmin_i,
                                const int* __restrict__ dmax_i,
                                float* __restrict__ dmin_f,
                                float* __restrict__ denom_f) {
    int b = threadIdx.x;
    if (b >= NB) return;
    float mn = __int_as_float(dmin_i[b]);
    float mx = __int_as_float(dmax_i[b]);
    if (!(mn < INFINITY)) mn = 0.0f;   // stayed at +inf -> 0
    if (!(mx > -INFINITY)) mx = 1.0f;  // stayed at -inf -> 1
    const float den = (mx > mn) ? (mx - mn) : 1.0f;
    dmin_f[b] = mn;
    denom_f[b] = den;
}

// ---------------------------------------------------------------------------
// Kernel 5: out[i][j][h] = tanh(tf[i][h] - df[j][h] + dnorm(i,j)) on the
// block-diagonal mask, exact 0 elsewhere. One wave per (i,j), lane -> 4 h's.
// Nontemporal stores: 524 MB streamed once, never re-read.
// ---------------------------------------------------------------------------
__global__ void imap_kernel(const float* __restrict__ tf,
                            const float* __restrict__ df,
                            const float* __restrict__ tpos,
                            const float* __restrict__ dpos,
                            const int* __restrict__ seg_res,
                            const int* __restrict__ seg_atom,
                            const float* __restrict__ dmin_f,
                            const float* __restrict__ denom_f,
                            float* __restrict__ out) {
    const int lane = threadIdx.x & 31;
    const int wave = threadIdx.x >> 5;
    const int i = blockIdx.y;
    const int j = blockIdx.x * 8 + wave;

    const int br = seg_res[i];
    const int ba = seg_atom[j];

    const v4f* tf4 = (const v4f*)(tf + (size_t)i * H);
    v4f* outp = (v4f*)(out + ((size_t)i * NA + j) * H) + lane;

    v4f o;
    if (br == ba) {  // wave-uniform branch
        const v4f t = tf4[lane];
        const v4f d = ((const v4f*)(df + (size_t)j * H))[lane];
        const float dx = tpos[3 * i + 0] - dpos[3 * j + 0];
        const float dy = tpos[3 * i + 1] - dpos[3 * j + 1];
        const float dz = tpos[3 * i + 2] - dpos[3 * j + 2];
        const float dist = sqrtf(dx * dx + dy * dy + dz * dz);
        const float dn = (dist - dmin_f[br]) / denom_f[br];
        o.x = tanhf(t.x - d.x + dn);
        o.y = tanhf(t.y - d.y + dn);
        o.z = tanhf(t.z - d.z + dn);
        o.w = tanhf(t.w - d.w + dn);
    } else {
        o = (v4f){0.0f, 0.0f, 0.0f, 0.0f};
    }
    __builtin_nontemporal_store(o, outp);
}

// ---------------------------------------------------------------------------
extern "C" void kernel_launch(void* const* d_in, const int* in_sizes, int n_in,
                              void* d_out, int out_size, void* d_ws, size_t ws_size,
                              hipStream_t stream) {
    const float* target_feature = (const float*)d_in[0];  // [NR, TD]
    const float* drug_feature   = (const float*)d_in[1];  // [NA, DD]
    const float* target_pos     = (const float*)d_in[2];  // [NR, 3]
    const float* drug_pos       = (const float*)d_in[3];  // [NA, 3]
    const float* Wt             = (const float*)d_in[4];  // [TD, H]
    const float* bt             = (const float*)d_in[5];  // [H]
    const float* Wd             = (const float*)d_in[6];  // [DD, H]
    const float* bd             = (const float*)d_in[7];  // [H]
    const int*   seg_res        = (const int*)d_in[8];    // [NR]
    const int*   seg_atom       = (const int*)d_in[9];    // [NA]
    float* out = (float*)d_out;

    // Workspace layout (~1.8 MB)
    float* ws     = (float*)d_ws;
    float* tf     = ws;                   // NR*H
    float* df     = tf + (size_t)NR * H;  // NA*H
    int*   dmin_i = (int*)(df + (size_t)NA * H);
    int*   dmax_i = dmin_i + NB;
    float* dmin_f = (float*)(dmax_i + NB);
    float* denom_f = dmin_f + NB;

    (void)in_sizes; (void)n_in; (void)out_size; (void)ws_size;

    // 1. init sentinels (deterministic across graph replays)
    init_minmax_kernel<<<1, 32, 0, stream>>>(dmin_i, dmax_i);

    // 2. WMMA projections: tf = target_feature @ Wt + bt ; df = drug @ Wd + bd
    {
        int tiles = (NR / 16) * (H / 16);  // 1600 -> 200 blocks of 8 waves
        gemm_wmma_f32_kernel<<<tiles / 8, 256, 0, stream>>>(target_feature, Wt, bt, tf, NR, H, TD);
    }
    {
        int tiles = (NA / 16) * (H / 16);  // 160 -> 20 blocks
        gemm_wmma_f32_kernel<<<tiles / 8, 256, 0, stream>>>(drug_feature, Wd, bd, df, NA, H, DD);
    }

    // 3. masked per-batch distance min/max
    minmax_kernel<<<(NR + 255) / 256, 256, 0, stream>>>(target_pos, drug_pos,
                                                        seg_res, seg_atom, dmin_i, dmax_i);

    // 4. finalize dmin / denom
    finalize_kernel<<<1, 32, 0, stream>>>(dmin_i, dmax_i, dmin_f, denom_f);

    // 5. big streaming map: grid (NA/8, NR), 8 waves/block, one wave per (i,j)
    dim3 grid(NA / 8, NR);
    imap_kernel<<<grid, 256, 0, stream>>>(tf, df, target_pos, drug_pos,
                                          seg_res, seg_atom, dmin_f, denom_f, out);
}